// SlotAttention_16836271800414
// MI455X (gfx1250) — compile-verified
//
#include <hip/hip_runtime.h>
#include <math.h>

#define B_ 32
#define N_ 4096
#define HID_ 128
#define K_ 7
#define D_ 128
#define H_ 4
#define DH_ 32
#define MLPH_ 256
#define ITERS_ 3
#define LN_EPS 1e-5f
#define EPS_ 1e-8f
#define SCALE_ 0.17677669529663687f  /* 32^-0.5 */

typedef float v2f __attribute__((ext_vector_type(2)));
typedef float v8f __attribute__((ext_vector_type(8)));

// D = A(16x4 f32) * B(4x16 f32) + C(16x16 f32)
#define WMMA_F32(a, b, c) \
  __builtin_amdgcn_wmma_f32_16x16x4_f32(false, (a), false, (b), (short)0, (c), false, false)

// ---------------------------------------------------------------------------
// K1: slots = mu + sigma * noise   ([B,K,D] -> d_out)
// ---------------------------------------------------------------------------
__global__ void k_init_slots(const float* __restrict__ noise,
                             const float* __restrict__ mu,
                             const float* __restrict__ sigma,
                             float* __restrict__ slots) {
  int i = blockIdx.x * 256 + threadIdx.x;
  if (i < B_ * K_ * D_) {
    int d = i & (D_ - 1);
    slots[i] = mu[d] + sigma[d] * noise[i];
  }
}

// store one 16x16 C tile into [B,H,N,DH] layout
__device__ __forceinline__ void store_tile(float* __restrict__ dst, const v8f& c,
                                           int row0, int m0, int n0, int half,
                                           int l15) {
  for (int j = 0; j < 8; ++j) {
    int r = m0 + j + 8 * half;
    int grow = row0 + r;
    int bidx = grow >> 12, nn = grow & (N_ - 1);
    int d = n0 + l15, h = d >> 5, dh = d & 31;
    dst[((((size_t)bidx * H_ + h) * N_) + nn) * DH_ + dh] = c[j];
  }
}

// ---------------------------------------------------------------------------
// K2: x_ln = LN(inputs); k = x_ln@Wk+bk; v = x_ln@Wv+bv, stored [B,H,N,DH]
// 64-row tile per block, 8 waves: 4 waves per {K,V} matrix, each wave owns a
// 32x64 block, walked as two 32x32 macro-tiles -> 2M x 2N register blocking (4 WMMAs per operand fetch).
// ---------------------------------------------------------------------------
__global__ __launch_bounds__(256) void k_ln_kv(
    const float* __restrict__ x, const float* __restrict__ g,
    const float* __restrict__ bln,
    const float* __restrict__ Wk, const float* __restrict__ bk,
    const float* __restrict__ Wv, const float* __restrict__ bvec,
    float* __restrict__ kws, float* __restrict__ vws) {
  __shared__ float xt[64 * 132];  // padded stride -> conflict-free strided reads
  const int tid = threadIdx.x;
  const int row0 = blockIdx.x * 64;  // global row index into B*N

  // coalesced float4 load of 64x128 tile
  const float4* xin = (const float4*)(x + (size_t)row0 * HID_);
  for (int i = tid; i < 64 * 32; i += 256) {
    int r = i >> 5, c4 = (i & 31) * 4;
    float4 vv = xin[i];
    xt[r * 132 + c4 + 0] = vv.x;
    xt[r * 132 + c4 + 1] = vv.y;
    xt[r * 132 + c4 + 2] = vv.z;
    xt[r * 132 + c4 + 3] = vv.w;
  }
  __syncthreads();

  // LayerNorm per row (threads 0..63)
  if (tid < 64) {
    float* rp = &xt[tid * 132];
    float m = 0.f;
    for (int c = 0; c < 128; ++c) m += rp[c];
    m *= (1.0f / 128.0f);
    float vv = 0.f;
    for (int c = 0; c < 128; ++c) { float d = rp[c] - m; vv += d * d; }
    vv *= (1.0f / 128.0f);
    float inv = rsqrtf(vv + LN_EPS);
    for (int c = 0; c < 128; ++c) rp[c] = (rp[c] - m) * inv * g[c] + bln[c];
  }
  __syncthreads();

  const int lane = tid & 31, wv = tid >> 5;
  const int half = lane >> 4, l15 = lane & 15;
  const int mat = wv & 1;    // 0 -> K, 1 -> V
  const int wid = wv >> 1;   // 0..3 within matrix
  const int mbase = (wid & 1) * 32;   // two M tiles: mbase, mbase+16
  const int nquad = (wid >> 1) * 64;  // four N tiles, walked as two pairs
  const float* W = mat ? Wv : Wk;
  const float* bias = mat ? bvec : bk;
  float* dst = mat ? vws : kws;

  for (int np = 0; np < 2; ++np) {
    const int n0 = nquad + np * 32;  // N-tile pair: n0, n0+16
    v8f c00, c01, c10, c11;
    float bb0 = bias[n0 + l15];
    float bb1 = bias[n0 + 16 + l15];
    for (int j = 0; j < 8; ++j) {
      c00[j] = bb0; c10[j] = bb0;  // bias depends on column only
      c01[j] = bb1; c11[j] = bb1;
    }
    for (int kb = 0; kb < 128; kb += 4) {
      int kk = kb + 2 * half;  // lanes 0-15: K 0,1 ; lanes 16-31: K 2,3
      v2f a0, a1, b0, b1;
      a0.x = xt[(mbase + l15) * 132 + kk];
      a0.y = xt[(mbase + l15) * 132 + kk + 1];
      a1.x = xt[(mbase + 16 + l15) * 132 + kk];
      a1.y = xt[(mbase + 16 + l15) * 132 + kk + 1];
      b0.x = W[kk * D_ + n0 + l15];
      b0.y = W[(kk + 1) * D_ + n0 + l15];
      b1.x = W[kk * D_ + n0 + 16 + l15];
      b1.y = W[(kk + 1) * D_ + n0 + 16 + l15];
      c00 = WMMA_F32(a0, b0, c00);
      c01 = WMMA_F32(a0, b1, c01);
      c10 = WMMA_F32(a1, b0, c10);
      c11 = WMMA_F32(a1, b1, c11);
    }
    store_tile(dst, c00, row0, mbase, n0, half, l15);
    store_tile(dst, c01, row0, mbase, n0 + 16, half, l15);
    store_tile(dst, c10, row0, mbase + 16, n0, half, l15);
    store_tile(dst, c11, row0, mbase + 16, n0 + 16, half, l15);
  }
}

// ---------------------------------------------------------------------------
// K3: sn = LN(slots); q = sn@Wq+bq -> [B,H,K,DH]   (tiny; one block per b)
// ---------------------------------------------------------------------------
__global__ __launch_bounds__(128) void k_slots_q(
    const float* __restrict__ slots, const float* __restrict__ g,
    const float* __restrict__ bln, const float* __restrict__ Wq,
    const float* __restrict__ bq, float* __restrict__ qws) {
  __shared__ float sn[K_ * D_];
  const int bidx = blockIdx.x, t = threadIdx.x;
  for (int i = t; i < K_ * D_; i += 128) sn[i] = slots[bidx * K_ * D_ + i];
  __syncthreads();
  if (t < K_) {
    float* rp = &sn[t * D_];
    float m = 0.f;
    for (int c = 0; c < D_; ++c) m += rp[c];
    m *= (1.f / D_);
    float vv = 0.f;
    for (int c = 0; c < D_; ++c) { float d = rp[c] - m; vv += d * d; }
    vv *= (1.f / D_);
    float inv = rsqrtf(vv + LN_EPS);
    for (int c = 0; c < D_; ++c) rp[c] = (rp[c] - m) * inv * g[c] + bln[c];
  }
  __syncthreads();
  const int d = t, h = d >> 5, dh = d & 31;
  for (int k = 0; k < K_; ++k) {
    float acc = bq[d];
    for (int m = 0; m < D_; ++m) acc += sn[k * D_ + m] * Wq[m * D_ + d];
    qws[((bidx * H_ + h) * K_ + k) * DH_ + dh] = acc;
  }
}

// ---------------------------------------------------------------------------
// K4: per (b,h): dots = q@k^T * SCALE; w = softmax over slots + EPS;
//     updates = (w @ v) / rowsum(w)   -- renorm folded into the division.
//     w@v accumulated with WMMA f32 16x16x4; each wave owns a 16-n chunk and
//     computes BOTH 16-wide d tiles from a single A fetch.
// ---------------------------------------------------------------------------
__global__ __launch_bounds__(256) void k_attn(
    const float* __restrict__ kws, const float* __restrict__ vws,
    const float* __restrict__ qws, float* __restrict__ upd) {
  __shared__ float qs[K_ * DH_];
  __shared__ float dots[K_ * 128];
  __shared__ float wsm[16 * 132];   // padded; rows 7..15 stay zero
  __shared__ float denW[4][K_];
  __shared__ float red[16 * 8 * 32];  // 8 waves x 2 d-tiles x v8f x 32 lanes
  const int bh = blockIdx.x;  // b*H + h
  const int bb = bh >> 2, h = bh & 3;
  const int t = threadIdx.x, lane = t & 31, wv = t >> 5;
  const int half = lane >> 4, l15 = lane & 15;

  for (int i = t; i < K_ * DH_; i += 256) qs[i] = qws[bh * K_ * DH_ + i];
  for (int i = t; i < 16 * 132; i += 256) wsm[i] = 0.f;
  if (t < 4 * K_) denW[t / K_][t % K_] = 0.f;

  v8f c0, c1;  // d-tiles 0..15 and 16..31
  for (int j = 0; j < 8; ++j) { c0[j] = 0.f; c1[j] = 0.f; }
  const float* kbase = kws + (size_t)bh * N_ * DH_;
  const float* vbase = vws + (size_t)bh * N_ * DH_;
  __syncthreads();

  for (int nt = 0; nt < N_ / 128; ++nt) {
    const int ntile = nt * 128;
    // ---- dots[k][n] for 128 n (2 threads per n split the 7 slots) ----
    {
      int n = t & 127, kh = t >> 7;
      const float4* kv4 = (const float4*)(kbase + (size_t)(ntile + n) * DH_);
      float kv[32];
      for (int i = 0; i < 8; ++i) {
        float4 f = kv4[i];
        kv[4 * i] = f.x; kv[4 * i + 1] = f.y; kv[4 * i + 2] = f.z; kv[4 * i + 3] = f.w;
      }
      // prefetch next tile's k (kh==0) / v (kh==1) rows into cache
      if (nt + 1 < N_ / 128) {
        const float* pf = (kh ? vbase : kbase) + (size_t)(ntile + 128 + n) * DH_;
        __builtin_prefetch(pf, 0, 0);
      }
      int k0 = kh ? 4 : 0, k1 = kh ? 7 : 4;
      for (int kk = k0; kk < k1; ++kk) {
        float acc = 0.f;
        const float* qp = &qs[kk * DH_];
        for (int d = 0; d < 32; ++d) acc += qp[d] * kv[d];
        dots[kk * 128 + n] = acc * SCALE_;
      }
    }
    __syncthreads();
    // ---- softmax over the 7 slots per n, +EPS; wave-partial row sums ----
    if (t < 128) {
      int n = t;
      float v0[K_];
      float mx = -1e30f;
      for (int kk = 0; kk < K_; ++kk) { v0[kk] = dots[kk * 128 + n]; mx = fmaxf(mx, v0[kk]); }
      float s = 0.f;
      for (int kk = 0; kk < K_; ++kk) { v0[kk] = __expf(v0[kk] - mx); s += v0[kk]; }
      float inv = 1.0f / s;
      for (int kk = 0; kk < K_; ++kk) {
        float wval = v0[kk] * inv + EPS_;
        wsm[kk * 132 + n] = wval;
        float r = wval;
        for (int off = 16; off > 0; off >>= 1) r += __shfl_xor(r, off, 32);
        if (lane == 0) denW[wv][kk] += r;  // waves 0..3, distinct rows
      }
    }
    __syncthreads();
    // ---- WMMA: c[16 slots x 32 d] += w[16 x 4n] * v[4n x 32d], A reused ----
    for (int s4 = 0; s4 < 4; ++s4) {
      int nb = wv * 16 + s4 * 4 + 2 * half;  // wave-private 16-n chunk
      v2f a, b0, b1;
      a.x = wsm[l15 * 132 + nb];
      a.y = wsm[l15 * 132 + nb + 1];
      const float* vp = vbase + (size_t)(ntile + nb) * DH_ + l15;
      b0.x = vp[0];
      b0.y = vp[DH_];
      b1.x = vp[16];
      b1.y = vp[DH_ + 16];
      c0 = WMMA_F32(a, b0, c0);
      c1 = WMMA_F32(a, b1, c1);
    }
    __syncthreads();
  }

  // ---- reduce the 8 n-chunk partials per d-tile, divide by den, store ----
  for (int j = 0; j < 8; ++j) {
    red[(wv * 2 + 0) * 256 + j * 32 + lane] = c0[j];
    red[(wv * 2 + 1) * 256 + j * 32 + lane] = c1[j];
  }
  __syncthreads();
  if (wv < 2) {  // wave 0 -> d-tile 0, wave 1 -> d-tile 1
    for (int j = 0; j < 8; ++j) {
      float s = 0.f;
      for (int w = 0; w < 8; ++w) s += red[(w * 2 + wv) * 256 + j * 32 + lane];
      int row = j + 8 * half;
      if (row < K_) {
        float den = denW[0][row] + denW[1][row] + denW[2][row] + denW[3][row];
        upd[((bb * K_ + row) * D_) + h * DH_ + wv * 16 + l15] = s / den;
      }
    }
  }
}

// ---------------------------------------------------------------------------
// K5: slots = GRU(updates, slots); slots += MLP(LN(slots))  (one block/slot)
// ---------------------------------------------------------------------------
__global__ __launch_bounds__(128) void k_gru_mlp(
    const float* __restrict__ upd, float* __restrict__ slots,
    const float* __restrict__ w_ih, const float* __restrict__ b_ih,
    const float* __restrict__ w_hh, const float* __restrict__ b_hh,
    const float* __restrict__ lng, const float* __restrict__ lnb,
    const float* __restrict__ W1, const float* __restrict__ b1,
    const float* __restrict__ W2, const float* __restrict__ b2) {
  __shared__ float xr[D_], hr[D_], lnr[D_], hid[MLPH_], red[4];
  const int bk = blockIdx.x, t = threadIdx.x;
  xr[t] = upd[bk * D_ + t];
  hr[t] = slots[bk * D_ + t];
  __syncthreads();

  float gi[3], gh[3];
  for (int gidx = 0; gidx < 3; ++gidx) {
    int j = gidx * D_ + t;
    float a = b_ih[j], c2 = b_hh[j];
    const float* wi = &w_ih[j * D_];
    const float* wh = &w_hh[j * D_];
    for (int m = 0; m < D_; ++m) { a += xr[m] * wi[m]; c2 += hr[m] * wh[m]; }
    gi[gidx] = a; gh[gidx] = c2;
  }
  float r = 1.f / (1.f + __expf(-(gi[0] + gh[0])));
  float z = 1.f / (1.f + __expf(-(gi[1] + gh[1])));
  float nn = tanhf(gi[2] + r * gh[2]);
  float s = (1.f - z) * nn + z * hr[t];

  // LayerNorm over the 128 s values
  float val = s;
  for (int off = 16; off > 0; off >>= 1) val += __shfl_xor(val, off, 32);
  if ((t & 31) == 0) red[t >> 5] = val;
  __syncthreads();
  float mean = (red[0] + red[1] + red[2] + red[3]) * (1.f / 128.f);
  float dv = s - mean;
  float v2 = dv * dv;
  for (int off = 16; off > 0; off >>= 1) v2 += __shfl_xor(v2, off, 32);
  __syncthreads();
  if ((t & 31) == 0) red[t >> 5] = v2;
  __syncthreads();
  float var = (red[0] + red[1] + red[2] + red[3]) * (1.f / 128.f);
  lnr[t] = dv * rsqrtf(var + LN_EPS) * lng[t] + lnb[t];
  __syncthreads();

  // hidden = relu(ln @ W1 + b1)  (each thread: cols t and t+128)
  for (int gidx = 0; gidx < 2; ++gidx) {
    int j = gidx * D_ + t;
    float a = b1[j];
    for (int m = 0; m < D_; ++m) a += lnr[m] * W1[m * MLPH_ + j];
    hid[j] = fmaxf(a, 0.f);
  }
  __syncthreads();

  float o = s + b2[t];
  for (int m = 0; m < MLPH_; ++m) o += hid[m] * W2[m * D_ + t];
  slots[bk * D_ + t] = o;
}

// ---------------------------------------------------------------------------
extern "C" void kernel_launch(void* const* d_in, const int* in_sizes, int n_in,
                              void* d_out, int out_size, void* d_ws, size_t ws_size,
                              hipStream_t stream) {
  const float* inputs  = (const float*)d_in[0];
  const float* noise   = (const float*)d_in[1];
  const float* mu      = (const float*)d_in[2];
  const float* sigma   = (const float*)d_in[3];
  const float* ln_in_g = (const float*)d_in[4];
  const float* ln_in_b = (const float*)d_in[5];
  const float* ln_s_g  = (const float*)d_in[6];
  const float* ln_s_b  = (const float*)d_in[7];
  const float* ln_m_g  = (const float*)d_in[8];
  const float* ln_m_b  = (const float*)d_in[9];
  const float* Wq  = (const float*)d_in[10];
  const float* bq  = (const float*)d_in[11];
  const float* Wk  = (const float*)d_in[12];
  const float* bk  = (const float*)d_in[13];
  const float* Wv  = (const float*)d_in[14];
  const float* bv  = (const float*)d_in[15];
  const float* wih = (const float*)d_in[16];
  const float* bih = (const float*)d_in[17];
  const float* whh = (const float*)d_in[18];
  const float* bhh = (const float*)d_in[19];
  const float* W1  = (const float*)d_in[20];
  const float* b1  = (const float*)d_in[21];
  const float* W2  = (const float*)d_in[22];
  const float* b2  = (const float*)d_in[23];

  float* slots = (float*)d_out;               // slots live in d_out throughout
  float* ws = (float*)d_ws;
  float* kws = ws;                                        // [B,H,N,DH]
  float* vws = kws + (size_t)B_ * H_ * N_ * DH_;          // [B,H,N,DH]
  float* qws = vws + (size_t)B_ * H_ * N_ * DH_;          // [B,H,K,DH]
  float* upd = qws + (size_t)B_ * H_ * K_ * DH_;          // [B,K,D]

  k_init_slots<<<(B_ * K_ * D_ + 255) / 256, 256, 0, stream>>>(noise, mu, sigma, slots);
  k_ln_kv<<<(B_ * N_) / 64, 256, 0, stream>>>(inputs, ln_in_g, ln_in_b,
                                              Wk, bk, Wv, bv, kws, vws);
  for (int it = 0; it < ITERS_; ++it) {
    k_slots_q<<<B_, 128, 0, stream>>>(slots, ln_s_g, ln_s_b, Wq, bq, qws);
    k_attn<<<B_ * H_, 256, 0, stream>>>(kws, vws, qws, upd);
    k_gru_mlp<<<B_ * K_, 128, 0, stream>>>(upd, slots, wih, bih, whh, bhh,
                                           ln_m_g, ln_m_b, W1, b1, W2, b2);
  }
}